// Net_41807211659639
// MI455X (gfx1250) — compile-verified
//
#include <hip/hip_runtime.h>
#include <hip/hip_bf16.h>

typedef float v2f __attribute__((ext_vector_type(2)));
typedef float v8f __attribute__((ext_vector_type(8)));

// ---------------------------------------------------------------------------
// WMMA f32 GEMM: C[M x Nn] = A[M x K] @ B[K x Nn],  K = KC (compile time)
// 256-thread block = 8 waves; each wave owns one 16-row M-tile and all NT
// 16x16 N-tiles (so A streams from HBM exactly once).  The whole B panel
// (KC x NT*16, zero-padded past Nn) is staged in LDS once per block, so the
// inner loop is: 1x global_load_b64 (A) + 2*NT ds_load_b32 (B, conflict-free)
// + NT v_wmma_f32_16x16x4_f32 with f32 accumulate (full reference precision).
// A layout (ISA 7.12.2, 32-bit A 16x4): lanes 0-15 -> K=0,1 ; lanes 16-31 -> K=2,3
// C/D layout: VGPR r, lanes 0-15 -> M=r ; lanes 16-31 -> M=8+r ; N = lane&15
// ---------------------------------------------------------------------------
template <int NT, int KC>
__global__ __launch_bounds__(256) void gemm_wmma_f32(
    const float* __restrict__ A, const float* __restrict__ B,
    float* __restrict__ C, int M, int Nn, int lda, int ldb, int ldc) {
  constexpr int COLP = NT * 16;               // padded N (multiple of 16)
  __shared__ float sB[KC * COLP];

  const int tid = threadIdx.x;

  // ---- stage B panel into LDS (float4 chunks; Nn % 4 == 0 here) ----
  constexpr int C4 = COLP / 4;
  for (int i = tid; i < KC * C4; i += 256) {
    const int k  = i / C4;
    const int c4 = (i - k * C4) * 4;
    float4 v = make_float4(0.f, 0.f, 0.f, 0.f);
    if (c4 < Nn)
      v = *reinterpret_cast<const float4*>(B + (size_t)k * ldb + c4);
    *reinterpret_cast<float4*>(&sB[k * COLP + c4]) = v;
  }
  __syncthreads();

  const int wave  = tid >> 5;
  const int lane  = tid & 31;
  const int mtile = blockIdx.x * 8 + wave;

  if (mtile * 16 < M) {                        // wave-uniform: EXEC all-ones
    const int m0   = mtile * 16;
    const int rowA = min(m0 + (lane & 15), M - 1);
    const int koff = (lane >> 4) << 1;         // 0 for lanes 0-15, 2 for 16-31
    const int colL = lane & 15;
    const float* __restrict__ aRow = A + (size_t)rowA * lda;

    v8f acc[NT] = {};

#pragma unroll 2
    for (int kk = 0; kk < KC; kk += 4) {
      const float2 av =
          *reinterpret_cast<const float2*>(aRow + kk + koff);
      v2f a; a.x = av.x; a.y = av.y;

#pragma unroll
      for (int nt = 0; nt < NT; ++nt) {
        const int cb = nt * 16 + colL;
        v2f b;
        b.x = sB[(kk + koff)     * COLP + cb];
        b.y = sB[(kk + koff + 1) * COLP + cb];
        acc[nt] = __builtin_amdgcn_wmma_f32_16x16x4_f32(
            false, a, false, b, (short)0, acc[nt], false, false);
      }
    }

    // ---- epilogue: one predicate per N-tile, precomputed row base ----
    const int  rbase = m0 + ((lane >> 4) << 3);
    float* __restrict__ crow = C + (size_t)rbase * ldc;
    const bool fullM = (m0 + 16 <= M);         // wave-uniform (true here)

    if (fullM) {
#pragma unroll
      for (int nt = 0; nt < NT; ++nt) {
        const int cc = nt * 16 + colL;
        if (cc < Nn) {                         // single exec-mask toggle / tile
          float* __restrict__ cp = crow + cc;
#pragma unroll
          for (int r = 0; r < 8; ++r) cp[(size_t)r * ldc] = acc[nt][r];
        }
      }
    } else {                                    // M-tail (not taken for M%16==0)
#pragma unroll
      for (int nt = 0; nt < NT; ++nt) {
        const int cc = nt * 16 + colL;
#pragma unroll
        for (int r = 0; r < 8; ++r)
          if (cc < Nn && rbase + r < M) crow[(size_t)r * ldc + cc] = acc[nt][r];
      }
    }
  }
}

// ---------------------------------------------------------------------------
// GCN helpers
// ---------------------------------------------------------------------------
__global__ void fill_f32(float* __restrict__ p, float v, int n) {
  int i = blockIdx.x * blockDim.x + threadIdx.x;
  if (i < n) p[i] = v;
}

__global__ void deg_accum(const int* __restrict__ dst, float* __restrict__ deg,
                          int E) {
  int i = blockIdx.x * blockDim.x + threadIdx.x;
  if (i < E) atomicAdd(&deg[dst[i]], 1.0f);
}

__global__ void deg_finish(float* __restrict__ deg, int n) {
  int i = blockIdx.x * blockDim.x + threadIdx.x;
  if (i < n) deg[i] = rsqrtf(fmaxf(deg[i], 1.0f));
}

// norm[e] = dinv[src[e]] * dinv[dst[e]]  (shared by both conv layers)
__global__ void edge_norm(const int* __restrict__ src,
                          const int* __restrict__ dst,
                          const float* __restrict__ dinv,
                          float* __restrict__ norm, int E) {
  int e = blockIdx.x * blockDim.x + threadIdx.x;
  if (e < E) norm[e] = dinv[src[e]] * dinv[dst[e]];
}

// agg[n,f] = h[n,f] * dinv[n]^2   (self-loop contribution, norm = dinv*dinv)
__global__ void init_agg(const float* __restrict__ h,
                         const float* __restrict__ dinv,
                         float* __restrict__ agg, int N, int F) {
  int i = blockIdx.x * blockDim.x + threadIdx.x;
  if (i < N * F) {
    int n = i / F;
    float d = dinv[n];
    agg[i] = h[i] * d * d;
  }
}

// One thread per (edge, 4-feature chunk): agg[dst] += h[src] * norm[e]
__global__ void edge_agg(const float* __restrict__ h,
                         const float* __restrict__ norm,
                         const int* __restrict__ src,
                         const int* __restrict__ dst,
                         float* __restrict__ agg, int E, int F, int chunks) {
  int idx = blockIdx.x * blockDim.x + threadIdx.x;
  int e = idx / chunks;
  if (e >= E) return;
  int c  = idx - e * chunks;
  int s  = src[e];
  int d  = dst[e];
  float nrm = norm[e];
  int f0 = c * 4;
  const float4 hv =
      *reinterpret_cast<const float4*>(h + (size_t)s * F + f0);
  float* base = agg + (size_t)d * F + f0;
  atomicAdd(base + 0, hv.x * nrm);
  atomicAdd(base + 1, hv.y * nrm);
  atomicAdd(base + 2, hv.z * nrm);
  atomicAdd(base + 3, hv.w * nrm);
}

__global__ void bias_relu(const float* __restrict__ agg,
                          const float* __restrict__ b,
                          float* __restrict__ outp, int N, int F) {
  int i = blockIdx.x * blockDim.x + threadIdx.x;
  if (i < N * F) {
    int f = i % F;
    outp[i] = fmaxf(agg[i] + b[f], 0.0f);
  }
}

// emb[n] = relu(agg[n] + b2) / max(||.||, 1)   (F2 == 16)
__global__ void finalize_emb(const float* __restrict__ agg,
                             const float* __restrict__ b,
                             float* __restrict__ emb, int N) {
  int n = blockIdx.x * blockDim.x + threadIdx.x;
  if (n >= N) return;
  float v[16];
  const float4* ap = reinterpret_cast<const float4*>(agg + (size_t)n * 16);
#pragma unroll
  for (int q = 0; q < 4; ++q) {
    float4 t = ap[q];
    v[q * 4 + 0] = t.x; v[q * 4 + 1] = t.y;
    v[q * 4 + 2] = t.z; v[q * 4 + 3] = t.w;
  }
  float ss = 0.0f;
#pragma unroll
  for (int k = 0; k < 16; ++k) {
    v[k] = fmaxf(v[k] + b[k], 0.0f);
    ss += v[k] * v[k];
  }
  float sc = 1.0f / fmaxf(sqrtf(ss), 1.0f);
  float4* op = reinterpret_cast<float4*>(emb + (size_t)n * 16);
#pragma unroll
  for (int q = 0; q < 4; ++q) {
    float4 t;
    t.x = v[q * 4 + 0] * sc; t.y = v[q * 4 + 1] * sc;
    t.z = v[q * 4 + 2] * sc; t.w = v[q * 4 + 3] * sc;
    op[q] = t;
  }
}

// ---------------------------------------------------------------------------
// Edge MLP: feat = [ (emb_u - emb_v)^2 (16) , PI (25) ] (41)
//   t = leaky_relu(feat @ W1 + b1, 0.2); s = |t @ W + b|; clip; sigmoid(2-s)
// Weights staged into LDS (broadcast reads), one edge per thread.
// ---------------------------------------------------------------------------
__global__ __launch_bounds__(256) void edge_mlp(
    const float* __restrict__ emb, const int* __restrict__ tedges,
    const float* __restrict__ PI, const float* __restrict__ l1W,
    const float* __restrict__ l1b, const float* __restrict__ lW,
    const float* __restrict__ lb, float* __restrict__ out, int ED) {
  __shared__ float sW1[41 * 25];
  __shared__ float sb1[25];
  __shared__ float sW[25];
  __shared__ float sb0;
  const int tid = threadIdx.x;
  for (int i = tid; i < 41 * 25; i += 256) sW1[i] = l1W[i];
  if (tid < 25) { sb1[tid] = l1b[tid]; sW[tid] = lW[tid]; }
  if (tid == 0) sb0 = lb[0];
  __syncthreads();

  const int e = blockIdx.x * 256 + tid;
  if (e >= ED) return;

  const int u = tedges[2 * e + 0];
  const int v = tedges[2 * e + 1];
  float feat[41];
  const float4* eu = reinterpret_cast<const float4*>(emb + (size_t)u * 16);
  const float4* ev = reinterpret_cast<const float4*>(emb + (size_t)v * 16);
#pragma unroll
  for (int q = 0; q < 4; ++q) {
    float4 a = eu[q], b = ev[q];
    float dx = a.x - b.x, dy = a.y - b.y, dz = a.z - b.z, dw = a.w - b.w;
    feat[q * 4 + 0] = dx * dx; feat[q * 4 + 1] = dy * dy;
    feat[q * 4 + 2] = dz * dz; feat[q * 4 + 3] = dw * dw;
  }
  const float* pi = PI + (size_t)e * 25;
#pragma unroll
  for (int q = 0; q < 25; ++q) feat[16 + q] = pi[q];

  float s = 0.0f;
#pragma unroll 5
  for (int j = 0; j < 25; ++j) {
    float t = sb1[j];
#pragma unroll
    for (int k = 0; k < 41; ++k) t = fmaf(feat[k], sW1[k * 25 + j], t);
    t = t > 0.0f ? t : 0.2f * t;           // leaky_relu(0.2)
    s = fmaf(t, sW[j], s);
  }
  s = fabsf(s + sb0);
  s = fminf(fmaxf(s, 0.0f), 40.0f);
  out[e] = 1.0f / (1.0f + expf(s - 2.0f)); // sigmoid(2 - s)
}

// ---------------------------------------------------------------------------
// Launcher
// ---------------------------------------------------------------------------
extern "C" void kernel_launch(void* const* d_in, const int* in_sizes, int n_in,
                              void* d_out, int out_size, void* d_ws,
                              size_t ws_size, hipStream_t stream) {
  const float* x      = (const float*)d_in[0];
  const int*   eidx   = (const int*)d_in[1];
  const int*   tedges = (const int*)d_in[2];
  const float* PI     = (const float*)d_in[3];
  const float* W1     = (const float*)d_in[4];
  const float* b1     = (const float*)d_in[5];
  const float* W2     = (const float*)d_in[6];
  const float* b2     = (const float*)d_in[7];
  const float* l1W    = (const float*)d_in[8];
  const float* l1b    = (const float*)d_in[9];
  const float* lW     = (const float*)d_in[10];
  const float* lb     = (const float*)d_in[11];
  float* out = (float*)d_out;

  const int F_IN = 256, F1 = 100, F2 = 16;
  const int N  = in_sizes[0] / F_IN;
  const int E  = in_sizes[1] / 2;
  const int ED = in_sizes[2] / 2;
  const int* src = eidx;
  const int* dst = eidx + E;

  // Workspace carve (float units, 16B-aligned chunks)
  float* ws = (float*)d_ws;
  size_t off = 0;
  auto carve = [&](size_t cnt) {
    float* p = ws + off;
    off += (cnt + 3) & ~(size_t)3;
    return p;
  };
  float* dinv  = carve((size_t)N);
  float* enorm = carve((size_t)E);       // per-edge norm, shared by both convs
  float* h0    = carve((size_t)N * F1);  // x@W1, then reused as h1 (post-relu)
  float* agg1  = carve((size_t)N * F1);
  float* hh    = carve((size_t)N * F2);  // h1@W2, then reused as emb
  float* agg2  = carve((size_t)N * F2);
  float* h1  = h0;
  float* emb = hh;

  const int TB = 256;
  auto blocks = [](long long n, int tb) { return (unsigned)((n + tb - 1) / tb); };
  const unsigned mblocks = (unsigned)((N + 127) / 128);  // 8 M-tiles per block

  // --- degree / normalization ---
  fill_f32<<<blocks(N, TB), TB, 0, stream>>>(dinv, 1.0f, N);      // self loop
  deg_accum<<<blocks(E, TB), TB, 0, stream>>>(dst, dinv, E);
  deg_finish<<<blocks(N, TB), TB, 0, stream>>>(dinv, N);
  edge_norm<<<blocks(E, TB), TB, 0, stream>>>(src, dst, dinv, enorm, E);

  // --- conv1: h0 = x @ W1 ; aggregate ; h1 = relu(agg + b1) ---
  gemm_wmma_f32<7, 256><<<mblocks, 256, 0, stream>>>(
      x, W1, h0, N, F1, F_IN, F1, F1);
  init_agg<<<blocks((long long)N * F1, TB), TB, 0, stream>>>(h0, dinv, agg1, N, F1);
  edge_agg<<<blocks((long long)E * (F1 / 4), TB), TB, 0, stream>>>(
      h0, enorm, src, dst, agg1, E, F1, F1 / 4);
  bias_relu<<<blocks((long long)N * F1, TB), TB, 0, stream>>>(agg1, b1, h1, N, F1);

  // --- conv2: hh = h1 @ W2 ; aggregate ; emb = normalize(relu(agg + b2)) ---
  gemm_wmma_f32<1, 100><<<mblocks, 256, 0, stream>>>(
      h1, W2, hh, N, F2, F1, F2, F2);
  init_agg<<<blocks((long long)N * F2, TB), TB, 0, stream>>>(hh, dinv, agg2, N, F2);
  edge_agg<<<blocks((long long)E * (F2 / 4), TB), TB, 0, stream>>>(
      hh, enorm, src, dst, agg2, E, F2, F2 / 4);
  finalize_emb<<<blocks(N, TB), TB, 0, stream>>>(agg2, b2, emb, N);

  // --- edge MLP -> probabilities ---
  edge_mlp<<<blocks(ED, TB), TB, 0, stream>>>(
      emb, tedges, PI, l1W, l1b, lW, lb, out, ED);
}